// _Gaussian2d_77275051590149
// MI455X (gfx1250) — compile-verified
//
#include <hip/hip_runtime.h>

// Depthwise 5x5 Gaussian conv, NCHW fp32, B=16 C=64 H=W=256, SAME zero padding.
// Roofline: 537 MB HBM traffic @ 23.3 TB/s ~= 23 us floor; separable compute
// (0.84 G FMA) and LDS traffic both sit below that, so the kernel is designed
// around the CDNA5 async global->LDS path:
//  - 64x64 output tile per block; halo staged as 68 rows x 72 cols (16B
//    aligned) in LDS via global_load_async_to_lds_b128 (ASYNCcnt), then
//    s_wait_asynccnt 0 + barrier.
//  - Gaussian prior is exactly rank-1 (w = outer(u,u)); u recovered from row
//    sums -> 10 FMA per staged row per column instead of 25.
//  - Each thread produces a 1x16 column strip; each staged LDS row is read
//    once (5 taps) and the horizontal dot h is reused across 5 vertical taps.
//  - Output stores: 32 lanes of a wave write 32 consecutive floats -> each
//    store covers one 128B line.

#define H_      256
#define W_      256
#define NCH     64
#define TILE    64
#define HROWS   (TILE + 4)          // 68 staged rows
#define HCOLS   72                  // staged cols (x0-4 .. x0+67), 16B aligned
#define NELEM   (HROWS * HCOLS)     // 4896
#define NCHUNK  (HROWS * (HCOLS/4)) // 68 * 18 = 1224 b128 chunks

__global__ __launch_bounds__(256)
void _gauss_dw5_kernel(const float* __restrict__ x,
                       const float* __restrict__ prior,
                       float* __restrict__ out) {
    __shared__ __align__(16) float tile[NELEM];

    const int tid    = threadIdx.x;
    const int tileid = blockIdx.x;           // 0..15 (4x4 tiles per plane)
    const int plane  = blockIdx.y;           // 0..1023 (= b*64 + c)
    const int c      = plane & (NCH - 1);
    const int x0     = (tileid & 3) * TILE;
    const int y0     = (tileid >> 2) * TILE;

    const float* src = x   + (size_t)plane * (H_ * W_);
    float*       dst = out + (size_t)plane * (H_ * W_);

    // Zero-fill only when the staged halo can fall outside the image.
    const bool border = (x0 == 0) | (y0 == 0) | (x0 + TILE == W_) | (y0 + TILE == H_);
    if (border) {
        for (int t = tid; t < NELEM; t += 256) tile[t] = 0.0f;
        __syncthreads();
    }

    // LDS byte offset of tile[0]: low 32 bits of the generic (flat) address.
    const unsigned lds_base = (unsigned)(unsigned long long)(void*)&tile[0];

    // Stage the halo with async global->LDS b128 loads: one 16B chunk per lane
    // per iteration. Chunk c of row ly covers global x = x0-4+4c .. x0-1+4c,
    // which is either fully inside or fully outside the image, so 16B-granular
    // predication + pre-zeroed LDS implements exact zero padding.
    for (int t = tid; t < NCHUNK; t += 256) {
        const int ly = t / (HCOLS / 4);
        const int ck = t - ly * (HCOLS / 4);
        const int gy = y0 - 2 + ly;
        const int gx = x0 - 4 + ck * 4;          // first of 4 consecutive floats
        if ((unsigned)gy < (unsigned)H_ && (unsigned)gx < (unsigned)W_) {
            const unsigned           lofs = lds_base + (unsigned)(ly * HCOLS + ck * 4) * 4u;
            const unsigned long long ga   =
                (unsigned long long)(const void*)(src + (gy * W_ + gx));
            asm volatile("global_load_async_to_lds_b128 %0, %1, off"
                         :: "v"(lofs), "v"(ga) : "memory");
        }
    }
    asm volatile("s_wait_asynccnt 0" ::: "memory");
    __syncthreads();

    // Recover the separable 1D factor from the (rank-1) 5x5 prior:
    //   w = outer(u, u)  =>  u[i] = rowsum(i) / sqrt(sum(w)).
    // Uniform per block -> scalar math / s_loads.
    const float* wp = prior + c * 25;
    float rs[5];
    float tot = 0.0f;
#pragma unroll
    for (int i = 0; i < 5; ++i) {
        float s = 0.0f;
#pragma unroll
        for (int j = 0; j < 5; ++j) s += wp[i * 5 + j];
        rs[i] = s;
        tot += s;
    }
    const float inv = rsqrtf(tot);
    float u[5];
#pragma unroll
    for (int i = 0; i < 5; ++i) u[i] = rs[i] * inv;

    // Each thread: column tx, output rows [ybase, ybase+16).
    const int tx    = tid & (TILE - 1);
    const int g     = tid >> 6;         // 0..3
    const int ybase = g * 16;

    float acc[16];
#pragma unroll
    for (int i = 0; i < 16; ++i) acc[i] = 0.0f;

    // Output col x0+tx maps to staged col tx+4-2 = tx+2 for the leftmost tap.
#pragma unroll
    for (int r = 0; r < 20; ++r) {
        const int base = (ybase + r) * HCOLS + tx + 2;
        const float a0 = tile[base + 0];
        const float a1 = tile[base + 1];
        const float a2 = tile[base + 2];
        const float a3 = tile[base + 3];
        const float a4 = tile[base + 4];
        // Horizontal 5-tap dot for this staged row.
        float h = u[0] * a0;
        h = fmaf(u[1], a1, h);
        h = fmaf(u[2], a2, h);
        h = fmaf(u[3], a3, h);
        h = fmaf(u[4], a4, h);
        // Scatter to the (up to 5) output rows that use this input row.
#pragma unroll
        for (int ky = 0; ky < 5; ++ky) {
            const int yo = r - ky;
            if (yo >= 0 && yo < 16) acc[yo] = fmaf(u[ky], h, acc[yo]);
        }
    }

#pragma unroll
    for (int i = 0; i < 16; ++i) {
        dst[(size_t)(y0 + ybase + i) * W_ + (x0 + tx)] = acc[i];
    }
}

extern "C" void kernel_launch(void* const* d_in, const int* in_sizes, int n_in,
                              void* d_out, int out_size, void* d_ws, size_t ws_size,
                              hipStream_t stream) {
    (void)in_sizes; (void)n_in; (void)out_size; (void)d_ws; (void)ws_size;
    const float* x     = (const float*)d_in[0];  // [16,64,256,256]
    const float* prior = (const float*)d_in[1];  // [64,1,5,5]
    float*       out   = (float*)d_out;          // [16,64,256,256]

    dim3 grid(16, 1024);  // 16 tiles/plane, 1024 planes
    _gauss_dw5_kernel<<<grid, 256, 0, stream>>>(x, prior, out);
}